// GraphTransformer_21105469292737
// MI455X (gfx1250) — compile-verified
//
#include <hip/hip_runtime.h>

// Graph transformer, restructured: per-NODE Q/K/V projection (not per-edge),
// fp32 WMMA (V_WMMA_F32_16X16X4_F32) for all dense GEMMs, edge phase via
// gather + f32 global atomics into L2-resident accumulators.

#define NN  100000      // nodes
#define NE  1600000     // edges
#define GIN 128         // gnn input dim
#define DM  64          // d_model == gnn output dim
#define NH  4           // heads (head dim 16)

typedef __attribute__((ext_vector_type(2))) float v2f;
typedef __attribute__((ext_vector_type(8))) float v8f;

static __device__ __forceinline__ v8f wmma_f32(v2f a, v2f b, v8f c) {
    // D = A(16x4 f32) * B(4x16 f32) + C(16x16 f32)
    return __builtin_amdgcn_wmma_f32_16x16x4_f32(
        /*neg_a=*/false, a, /*neg_b=*/false, b,
        /*c_mod=*/(short)0, c, /*reuse_a=*/false, /*reuse_b=*/false);
}

// ---------------------------------------------------------------------------
// embeds[n, d] = sum_k x[n,k] * Wp[d,k] + bp[d] + pos[n,d]
// One wave -> 16-row x 64-col strip.  A layout: lane l holds rows m=l%16,
// K-pair kb = k0 + 2*(l>>4).  C/D: VGPR j -> row j + 8*(l>>4), col l%16.
// ---------------------------------------------------------------------------
__global__ __launch_bounds__(256)
void gt_input_proj(const float* __restrict__ x, const float* __restrict__ Wp,
                   const float* __restrict__ bp, const float* __restrict__ pos,
                   float* __restrict__ emb) {
    const int lane = threadIdx.x & 31;
    const int wave = blockIdx.x * (blockDim.x >> 5) + (threadIdx.x >> 5);
    const int row0 = wave << 4;
    if (row0 >= NN) return;                      // wave-uniform: EXEC all-1s inside
    const int m  = lane & 15;
    const int kh = lane >> 4;

    v8f acc[4] = {};
    const float* xr = x + (size_t)(row0 + m) * GIN;
    for (int k0 = 0; k0 < GIN; k0 += 4) {
        const int kb = k0 + 2 * kh;
        v2f a; a.x = xr[kb]; a.y = xr[kb + 1];
        #pragma unroll
        for (int nt = 0; nt < 4; ++nt) {
            const int n = nt * 16 + m;
            v2f b;                               // B[k,n] = Wp[n,k] (Wp is [out,in])
            b.x = Wp[(size_t)n * GIN + kb];
            b.y = Wp[(size_t)n * GIN + kb + 1];
            acc[nt] = wmma_f32(a, b, acc[nt]);
        }
    }
    #pragma unroll
    for (int nt = 0; nt < 4; ++nt) {
        const int n = nt * 16 + m;
        const float bias = bp[n];
        #pragma unroll
        for (int j = 0; j < 8; ++j) {
            const size_t idx = (size_t)(row0 + j + 8 * kh) * DM + n;
            emb[idx] = acc[nt][j] + bias + pos[idx];
        }
    }
}

// ---------------------------------------------------------------------------
// Fused per-node Q/K/V projection: Q = emb @ qT, etc.  qT/kT/vT are [64,64]
// row-major indexed [k][n].  192 WMMAs per wave (12 per K-step).
// ---------------------------------------------------------------------------
__global__ __launch_bounds__(256)
void gt_qkv(const float* __restrict__ emb,
            const float* __restrict__ qT, const float* __restrict__ kT,
            const float* __restrict__ vT,
            float* __restrict__ Q, float* __restrict__ K, float* __restrict__ V) {
    const int lane = threadIdx.x & 31;
    const int wave = blockIdx.x * (blockDim.x >> 5) + (threadIdx.x >> 5);
    const int row0 = wave << 4;
    if (row0 >= NN) return;
    const int m  = lane & 15;
    const int kh = lane >> 4;

    v8f aq[4] = {}, ak[4] = {}, av[4] = {};
    const float* er = emb + (size_t)(row0 + m) * DM;
    for (int k0 = 0; k0 < DM; k0 += 4) {
        const int kb = k0 + 2 * kh;
        v2f a; a.x = er[kb]; a.y = er[kb + 1];
        #pragma unroll
        for (int nt = 0; nt < 4; ++nt) {
            const int n = nt * 16 + m;
            v2f b;
            b.x = qT[(size_t)kb * DM + n]; b.y = qT[(size_t)(kb + 1) * DM + n];
            aq[nt] = wmma_f32(a, b, aq[nt]);
            b.x = kT[(size_t)kb * DM + n]; b.y = kT[(size_t)(kb + 1) * DM + n];
            ak[nt] = wmma_f32(a, b, ak[nt]);
            b.x = vT[(size_t)kb * DM + n]; b.y = vT[(size_t)(kb + 1) * DM + n];
            av[nt] = wmma_f32(a, b, av[nt]);
        }
    }
    #pragma unroll
    for (int nt = 0; nt < 4; ++nt) {
        const int n = nt * 16 + m;
        #pragma unroll
        for (int j = 0; j < 8; ++j) {
            const size_t idx = (size_t)(row0 + j + 8 * kh) * DM + n;
            Q[idx] = aq[nt][j];
            K[idx] = ak[nt][j];
            V[idx] = av[nt][j];
        }
    }
}

// ---------------------------------------------------------------------------
// out[n, o] = sum_k emb[n,k] * invW[o,k] + invb[o]
// ---------------------------------------------------------------------------
__global__ __launch_bounds__(256)
void gt_out_proj(const float* __restrict__ emb, const float* __restrict__ invW,
                 const float* __restrict__ invb, float* __restrict__ out) {
    const int lane = threadIdx.x & 31;
    const int wave = blockIdx.x * (blockDim.x >> 5) + (threadIdx.x >> 5);
    const int row0 = wave << 4;
    if (row0 >= NN) return;
    const int m  = lane & 15;
    const int kh = lane >> 4;

    v8f acc[4] = {};
    const float* er = emb + (size_t)(row0 + m) * DM;
    for (int k0 = 0; k0 < DM; k0 += 4) {
        const int kb = k0 + 2 * kh;
        v2f a; a.x = er[kb]; a.y = er[kb + 1];
        #pragma unroll
        for (int nt = 0; nt < 4; ++nt) {
            const int n = nt * 16 + m;
            v2f b;                               // B[k,n] = invW[n,k]
            b.x = invW[(size_t)n * DM + kb];
            b.y = invW[(size_t)n * DM + kb + 1];
            acc[nt] = wmma_f32(a, b, acc[nt]);
        }
    }
    #pragma unroll
    for (int nt = 0; nt < 4; ++nt) {
        const int n = nt * 16 + m;
        const float bias = invb[n];
        #pragma unroll
        for (int j = 0; j < 8; ++j)
            out[(size_t)(row0 + j + 8 * kh) * DM + n] = acc[nt][j] + bias;
    }
}

// ---------------------------------------------------------------------------
// Edge pass 1: scores, exp, per-(dest,head) softmax denominators.
// ---------------------------------------------------------------------------
__global__ __launch_bounds__(256)
void gt_edge_score(const int* __restrict__ rows, const int* __restrict__ cols,
                   const float* __restrict__ Q, const float* __restrict__ K,
                   float* __restrict__ expS, float* __restrict__ norms) {
    const int e = blockIdx.x * blockDim.x + threadIdx.x;
    if (e >= NE) return;
    const int r = rows[e], c = cols[e];
    const float4* qr = (const float4*)(Q + (size_t)r * DM);
    const float4* kc = (const float4*)(K + (size_t)c * DM);
    float s[NH] = {0.f, 0.f, 0.f, 0.f};
    #pragma unroll
    for (int i = 0; i < 16; ++i) {               // head = i/4 (head dim 16)
        const float4 a = qr[i], b = kc[i];
        s[i >> 2] += a.x * b.x + a.y * b.y + a.z * b.z + a.w * b.w;
    }
    #pragma unroll
    for (int h = 0; h < NH; ++h) {
        const float sc = fminf(10.f, fmaxf(-10.f, s[h]));
        const float ex = expf(sc);
        expS[(size_t)e * NH + h] = ex;
        atomicAdd(norms + (size_t)r * NH + h, ex);
    }
}

// ---------------------------------------------------------------------------
// Edge pass 2: res[row] += (expS / norms[row]) * V[col]   (f32 atomics)
// ---------------------------------------------------------------------------
__global__ __launch_bounds__(256)
void gt_edge_agg(const int* __restrict__ rows, const int* __restrict__ cols,
                 const float* __restrict__ expS, const float* __restrict__ norms,
                 const float* __restrict__ V, float* __restrict__ res) {
    const int e = blockIdx.x * blockDim.x + threadIdx.x;
    if (e >= NE) return;
    const int r = rows[e], c = cols[e];
    float w[NH];
    #pragma unroll
    for (int h = 0; h < NH; ++h)
        w[h] = expS[(size_t)e * NH + h] / (norms[(size_t)r * NH + h] + 1e-8f);
    const float4* vc = (const float4*)(V + (size_t)c * DM);
    float* rr = res + (size_t)r * DM;
    #pragma unroll
    for (int i = 0; i < 16; ++i) {
        const float4 v = vc[i];
        const float wh = w[i >> 2];
        atomicAdd(rr + 4 * i + 0, wh * v.x);
        atomicAdd(rr + 4 * i + 1, wh * v.y);
        atomicAdd(rr + 4 * i + 2, wh * v.z);
        atomicAdd(rr + 4 * i + 3, wh * v.w);
    }
}

__global__ __launch_bounds__(256)
void gt_fill0(float* __restrict__ p, int n) {
    const int i = blockIdx.x * blockDim.x + threadIdx.x;
    if (i < n) p[i] = 0.f;
}

__global__ __launch_bounds__(256)
void gt_residual(float* __restrict__ emb, const float* __restrict__ res, int n) {
    const int i = blockIdx.x * blockDim.x + threadIdx.x;
    if (i < n) emb[i] += res[i];
}

// ---------------------------------------------------------------------------
extern "C" void kernel_launch(void* const* d_in, const int* in_sizes, int n_in,
                              void* d_out, int out_size, void* d_ws, size_t ws_size,
                              hipStream_t stream) {
    (void)in_sizes; (void)n_in; (void)out_size; (void)ws_size;

    const float* x    = (const float*)d_in[0];
    const int*   edge = (const int*)  d_in[1];     // [2, NE]: rows then cols
    const float* Wp   = (const float*)d_in[2];     // [64,128]
    const float* bp   = (const float*)d_in[3];     // [64]
    const float* pos  = (const float*)d_in[4];     // [NN,64]
    const float* qT   = (const float*)d_in[5];     // [2,64,64]
    const float* kT   = (const float*)d_in[6];
    const float* vT   = (const float*)d_in[7];
    const float* invW = (const float*)d_in[8];     // [64,64]
    const float* invb = (const float*)d_in[9];     // [64]
    float* out = (float*)d_out;

    const int* rows = edge;
    const int* cols = edge + NE;

    // workspace carve-up (fp32): 5*N*D + N*H + E*H floats (~155 MB)
    float* ws   = (float*)d_ws;
    float* emb  = ws;                      // N*D
    float* Q    = emb  + (size_t)NN * DM;  // N*D
    float* K    = Q    + (size_t)NN * DM;  // N*D
    float* V    = K    + (size_t)NN * DM;  // N*D
    float* res  = V    + (size_t)NN * DM;  // N*D   (res & norms contiguous -> one fill)
    float* norms= res  + (size_t)NN * DM;  // N*H
    float* expS = norms+ (size_t)NN * NH;  // E*H

    const int STRIPS = NN / 16;                          // 6250 (exact)
    const int WAVES_PER_BLK = 8;                         // 256 threads
    const int gemm_grid = (STRIPS + WAVES_PER_BLK - 1) / WAVES_PER_BLK;
    const int edge_grid = (NE + 255) / 256;
    const int nd = NN * DM;
    const int fill_n = NN * DM + NN * NH;                // res + norms
    const int fill_grid = (fill_n + 255) / 256;
    const int nd_grid = (nd + 255) / 256;

    gt_input_proj<<<gemm_grid, 256, 0, stream>>>(x, Wp, bp, pos, emb);

    for (int l = 0; l < 2; ++l) {
        const size_t off = (size_t)l * DM * DM;
        gt_qkv<<<gemm_grid, 256, 0, stream>>>(emb, qT + off, kT + off, vT + off, Q, K, V);
        gt_fill0<<<fill_grid, 256, 0, stream>>>(res, fill_n);
        gt_edge_score<<<edge_grid, 256, 0, stream>>>(rows, cols, Q, K, expS, norms);
        gt_edge_agg<<<edge_grid, 256, 0, stream>>>(rows, cols, expS, norms, V, res);
        gt_residual<<<nd_grid, 256, 0, stream>>>(emb, res, nd);
    }

    gt_out_proj<<<gemm_grid, 256, 0, stream>>>(emb, invW, invb, out);
}